// GCN_73581379715110
// MI455X (gfx1250) — compile-verified
//
#include <hip/hip_runtime.h>

#define N_NODES 100000
#define EMB_DIM 16
#define N_EDGES 3200000

typedef float v2f __attribute__((ext_vector_type(2)));
typedef float v8f __attribute__((ext_vector_type(8)));

// ---------------------------------------------------------------------------
// Kernel 1: zero the aggregation workspace (N_NODES * 16 floats, float4 wide)
// ---------------------------------------------------------------------------
__global__ void gcn_zero_agg(float4* __restrict__ agg4, int n4) {
    int i = blockIdx.x * blockDim.x + threadIdx.x;
    if (i < n4) agg4[i] = make_float4(0.f, 0.f, 0.f, 0.f);
}

// ---------------------------------------------------------------------------
// Kernel 2: edge scatter.  16 lanes cooperate on one edge: lane t handles
// feature dim t.  agg (6.4 MB) and emb (6.4 MB) live in the 192 MB L2, so
// the f32 atomics resolve at L2 atomic units; consecutive lanes write
// consecutive dims of the same destination row (64B-coalesced bursts).
// ---------------------------------------------------------------------------
__global__ void gcn_scatter(const float* __restrict__ emb,
                            const int*   __restrict__ src,
                            const int*   __restrict__ dst,
                            const float* __restrict__ w,
                            float*       __restrict__ agg) {
    long long gid = (long long)blockIdx.x * blockDim.x + threadIdx.x;
    int e = (int)(gid >> 4);
    int t = (int)(gid & 15);
    if (e >= N_EDGES) return;
    int s  = src[e];
    int d  = dst[e];
    float x = emb[s * EMB_DIM + t] * w[e];
    atomicAdd(&agg[d * EMB_DIM + t], x);
}

// ---------------------------------------------------------------------------
// Kernel 3: out = relu(agg @ W_rel.T + b_rel + emb @ W_root.T) via chained
// V_WMMA_F32_16X16X4_F32.  One wave32 owns a 16x16 output tile; K=16 is
// covered by 4 WMMA steps per weight matrix, accumulating in C/D.
//
// f32 A-matrix 16x4 layout (ISA 7.12.2): lane L<16 holds M=L, K={kb,kb+1};
// lane L+16 holds M=L, K={kb+2,kb+3}.  B 4x16 mirrors that with N striped
// across lanes.  C/D 16x16: VGPR r -> lanes 0-15: (M=r, N=lane),
// lanes 16-31: (M=r+8, N=lane-16).
// B[k][n] = W[n][k] implements the  X @ W.T  product.
// ---------------------------------------------------------------------------
__global__ __launch_bounds__(256)
void gcn_gemm(const float* __restrict__ agg,
              const float* __restrict__ emb,
              const float* __restrict__ W_rel,
              const float* __restrict__ b_rel,
              const float* __restrict__ W_root,
              float*       __restrict__ out) {
    int wave = (int)((blockIdx.x * blockDim.x + threadIdx.x) >> 5);
    int lane = threadIdx.x & 31;
    if (wave >= (N_NODES / 16)) return;   // wave-uniform: EXEC all-1s inside

    const int n   = lane & 15;   // output column / A row-within-tile
    const int hi  = lane >> 4;   // lane half selects K sub-pair
    const int m   = wave * 16 + n;

    // C init = bias broadcast per output column n
    const float bias = b_rel[n];
    v8f acc;
#pragma unroll
    for (int r = 0; r < 8; ++r) acc[r] = bias;

    const float* Arel  = agg + (long long)m * EMB_DIM;
    const float* Aroot = emb + (long long)m * EMB_DIM;

    // agg @ W_rel.T   (K = 0..15 in four 16x16x4 steps)
#pragma unroll
    for (int c = 0; c < 4; ++c) {
        const int kb = c * 4 + hi * 2;
        v2f a, b;
        a.x = Arel[kb];
        a.y = Arel[kb + 1];
        b.x = W_rel[n * EMB_DIM + kb];
        b.y = W_rel[n * EMB_DIM + kb + 1];
        acc = __builtin_amdgcn_wmma_f32_16x16x4_f32(
            /*neg_a=*/false, a, /*neg_b=*/false, b,
            /*c_mod=*/(short)0, acc, /*reuse_a=*/false, /*reuse_b=*/false);
    }

    // + emb @ W_root.T  (accumulate into same C/D)
#pragma unroll
    for (int c = 0; c < 4; ++c) {
        const int kb = c * 4 + hi * 2;
        v2f a, b;
        a.x = Aroot[kb];
        a.y = Aroot[kb + 1];
        b.x = W_root[n * EMB_DIM + kb];
        b.y = W_root[n * EMB_DIM + kb + 1];
        acc = __builtin_amdgcn_wmma_f32_16x16x4_f32(
            false, a, false, b, (short)0, acc, false, false);
    }

    // ReLU + store.  VGPR r holds row (wave*16 + r + hi*8), column n.
    // Lanes 0-31 of one VGPR cover two 64B-contiguous row segments.
#pragma unroll
    for (int r = 0; r < 8; ++r) {
        float v = acc[r];
        v = v > 0.f ? v : 0.f;
        out[(long long)(wave * 16 + r + hi * 8) * EMB_DIM + n] = v;
    }
}

// ---------------------------------------------------------------------------
extern "C" void kernel_launch(void* const* d_in, const int* in_sizes, int n_in,
                              void* d_out, int out_size, void* d_ws, size_t ws_size,
                              hipStream_t stream) {
    const float* emb        = (const float*)d_in[0];
    const float* W_rel      = (const float*)d_in[1];
    const float* b_rel      = (const float*)d_in[2];
    const float* W_root     = (const float*)d_in[3];
    const int*   edge_index = (const int*)  d_in[4];   // [2, E] flat
    const float* edge_w     = (const float*)d_in[5];
    float*       out        = (float*)d_out;

    float* agg = (float*)d_ws;                // N_NODES*16 floats = 6.4 MB

    const int* src = edge_index;              // row 0
    const int* dst = edge_index + N_EDGES;    // row 1

    // 1) zero agg
    int n4 = (N_NODES * EMB_DIM) / 4;
    gcn_zero_agg<<<(n4 + 255) / 256, 256, 0, stream>>>((float4*)agg, n4);

    // 2) scatter-sum over edges (16 lanes per edge)
    long long total = (long long)N_EDGES * 16;
    int sblocks = (int)((total + 255) / 256);
    gcn_scatter<<<sblocks, 256, 0, stream>>>(emb, src, dst, edge_w, agg);

    // 3) WMMA GEMM + bias + ReLU  (6250 waves, 8 waves/block)
    int ntiles = N_NODES / 16;                // 6250 exactly
    int wpb    = 256 / 32;
    int gblocks = (ntiles + wpb - 1) / wpb;
    gcn_gemm<<<gblocks, 256, 0, stream>>>(agg, emb, W_rel, b_rel, W_root, out);
}